// BSA_Encoder_88098369175783
// MI455X (gfx1250) — compile-verified
//
#include <hip/hip_runtime.h>
#include <cstdint>

#define T_LEN   8192
#define NSEQ    2048      // 32 batches * 64 channels
#define XCH     66
#define THRESH_F 0.679f

// ---------------------------------------------------------------------------
// Kernel A: per-(b,c) min/max over T, then origin = (x - mn) / (mx - mn).
// Fully coalesced float4 streaming; values kept in registers between passes.
// ---------------------------------------------------------------------------
__global__ __launch_bounds__(256) void minmax_norm_kernel(
    const float* __restrict__ x, float* __restrict__ out)
{
    const int g = blockIdx.x;              // sequence id 0..2047
    const int b = g >> 6, c = g & 63;
    const float4* row  = reinterpret_cast<const float4*>(
        x + (size_t)(b * XCH + c + 1) * T_LEN);
    float4* orow = reinterpret_cast<float4*>(
        out + (size_t)NSEQ * T_LEN + (size_t)g * T_LEN);
    const int tid = threadIdx.x;

    float4 v[8];
    v[0] = row[tid];
    float mn = fminf(fminf(v[0].x, v[0].y), fminf(v[0].z, v[0].w));
    float mx = fmaxf(fmaxf(v[0].x, v[0].y), fmaxf(v[0].z, v[0].w));
#pragma unroll
    for (int i = 1; i < 8; ++i) {
        v[i] = row[tid + i * 256];
        mn = fminf(mn, fminf(fminf(v[i].x, v[i].y), fminf(v[i].z, v[i].w)));
        mx = fmaxf(mx, fmaxf(fmaxf(v[i].x, v[i].y), fmaxf(v[i].z, v[i].w)));
    }
    // wave32 reduction
#pragma unroll
    for (int off = 16; off > 0; off >>= 1) {
        mn = fminf(mn, __shfl_xor(mn, off));
        mx = fmaxf(mx, __shfl_xor(mx, off));
    }
    __shared__ float smn[8], smx[8];
    const int wv = tid >> 5, ln = tid & 31;
    if (ln == 0) { smn[wv] = mn; smx[wv] = mx; }
    __syncthreads();
    mn = smn[0]; mx = smx[0];
#pragma unroll
    for (int i = 1; i < 8; ++i) { mn = fminf(mn, smn[i]); mx = fmaxf(mx, smx[i]); }

    const float den = mx - mn;
#pragma unroll
    for (int i = 0; i < 8; ++i) {
        float4 o;
        o.x = (v[i].x - mn) / den;
        o.y = (v[i].y - mn) / den;
        o.z = (v[i].z - mn) / den;
        o.w = (v[i].w - mn) / den;
        orow[tid + i * 256] = o;
    }
}

// ---------------------------------------------------------------------------
// Kernel B: BSA scan. One lane per sequence; 8192 serial steps.
// Origin stream staged to LDS via CDNA5 async global->LDS loads, double-
// buffered & pipelined with s_wait_asynccnt. 8 waves/block -> 2 waves/SIMD32
// so the second wave hides the per-step VALU dependency chain.
// decoded[t] = origin[t] - buf[0]_post  (identity: avoids a conv pass).
// ---------------------------------------------------------------------------
#define TT        64                 // timesteps per tile
#define NTILE     (T_LEN / TT)       // 128
#define ROWF      68                 // floats per lane row (64 + 4 pad, 16B aligned)
#define BUFSTRIDE (32 * ROWF)        // floats per buffer (per wave)
#define WPB       8                  // waves per block (256 threads)
// LDS: 8 waves * 2 buffers * 32 lanes * 68 floats * 4B = 139,264 bytes

// Copy one tile (TT floats per lane, 16 x b128) to LDS asynchronously.
__device__ __forceinline__ void async_tile16(unsigned lds_float_base,
                                             const float* gsrc)
{
    unsigned long long ga = (unsigned long long)(uintptr_t)gsrc;
    unsigned lb = lds_float_base * 4u;
#pragma unroll
    for (int c2 = 0; c2 < 16; ++c2) {
        asm volatile("global_load_async_to_lds_b128 %0, %1, off"
                     :: "v"(lb + (unsigned)(c2 * 16)),
                        "v"(ga + (unsigned long long)(c2 * 16))
                     : "memory");
    }
}

// One BSA step. Uses enclosing-scope bb[7] (residual buffer), r[8] (origin
// ring: r[i] == origin[t+i]), wf[7] (filter). Select fused with shift.
#define BSA_STEP(NEWV, OUTLV) do {                                          \
    float d0_ = bb[0] - wf[0], d1_ = bb[1] - wf[1], d2_ = bb[2] - wf[2];    \
    float d3_ = bb[3] - wf[3], d4_ = bb[4] - wf[4], d5_ = bb[5] - wf[5];    \
    float d6_ = bb[6] - wf[6];                                              \
    float e1_ = fabsf(d0_); e1_ += fabsf(d1_); e1_ += fabsf(d2_);           \
    e1_ += fabsf(d3_); e1_ += fabsf(d4_); e1_ += fabsf(d5_);                \
    e1_ += fabsf(d6_);                                                      \
    float e2_ = fabsf(bb[0]); e2_ += fabsf(bb[1]); e2_ += fabsf(bb[2]);     \
    e2_ += fabsf(bb[3]); e2_ += fabsf(bb[4]); e2_ += fabsf(bb[5]);          \
    e2_ += fabsf(bb[6]);                                                    \
    const bool sp_ = (e1_ <= e2_ - THRESH_F);                               \
    const float nb0_ = sp_ ? d0_ : bb[0];                                   \
    (OUTLV) = r[0] - nb0_;                                                  \
    bb[0] = sp_ ? d1_ : bb[1]; bb[1] = sp_ ? d2_ : bb[2];                   \
    bb[2] = sp_ ? d3_ : bb[3]; bb[3] = sp_ ? d4_ : bb[4];                   \
    bb[4] = sp_ ? d5_ : bb[5]; bb[5] = sp_ ? d6_ : bb[6];                   \
    bb[6] = r[7];                                                           \
    r[0] = r[1]; r[1] = r[2]; r[2] = r[3]; r[3] = r[4];                     \
    r[4] = r[5]; r[5] = r[6]; r[6] = r[7];                                  \
    r[7] = (NEWV);                                                          \
} while (0)

__global__ __launch_bounds__(256) void bsa_scan_kernel(
    const float* __restrict__ bw, float* __restrict__ out)
{
    __shared__ float lds[WPB * 2 * BUFSTRIDE];  // single shared object -> LDS offset 0

    const int tid  = threadIdx.x;
    const int wv   = tid >> 5;
    const int lane = tid & 31;
    const int g    = blockIdx.x * 256 + tid;    // sequence id
    const int ch   = g & 63;

    float wf[7];
#pragma unroll
    for (int j = 0; j < 7; ++j) wf[j] = bw[ch * 7 + j];

    const float* orow = out + (size_t)NSEQ * T_LEN + (size_t)g * T_LEN; // origin (kernel A)
    float*       drow = out + (size_t)g * T_LEN;                        // decoded

    const unsigned lanebase = (unsigned)(wv * 2 * BUFSTRIDE + lane * ROWF);
    const unsigned buf0 = lanebase;
    const unsigned buf1 = lanebase + BUFSTRIDE;

    // Prologue: prefetch tiles 0 and 1; <=16 guarantees tile 0 landed.
    async_tile16(buf0, orow);
    async_tile16(buf1, orow + TT);
    asm volatile("s_wait_asynccnt 16" ::: "memory");

    // Origin ring r[i] = origin[t+i]; residual buffer bb[j] init = origin[0..6].
    float r[8], bb[7];
#pragma unroll
    for (int i = 0; i < 8; ++i) r[i] = lds[buf0 + i];
#pragma unroll
    for (int j = 0; j < 7; ++j) bb[j] = r[j];

    // Pipeline invariant at top of iter k: tile k resident in B(k&1),
    // tile k+1 in flight (issued at end of iter k-1) or resident.
    for (int k = 0; k < NTILE; ++k) {
        const unsigned cur = (k & 1) ? buf1 : buf0;
        const unsigned nxt = (k & 1) ? buf0 : buf1;
        float* dtile = drow + k * TT;

        // Steps whose read-ahead (t+8) stays inside the current tile.
#pragma unroll
        for (int tt = 0; tt < TT - 8; ++tt) {
            BSA_STEP(lds[cur + tt + 8], dtile[tt]);
        }

        // Drain: tile k+1 (in flight for ~56 steps) is now resident.
        asm volatile("s_wait_asynccnt 0" ::: "memory");

        const bool have_next = (k + 1 < NTILE);
#pragma unroll
        for (int tt = TT - 8; tt < TT; ++tt) {
            float nv = have_next ? lds[nxt + (tt + 8 - TT)] : 0.0f; // origin[>=T]==0 pad
            BSA_STEP(nv, dtile[tt]);
        }

        // Issue tile k+2 into cur: cur is fully consumed, and next iter's
        // s_wait_asynccnt 0 drains it before iter k+2 reads it. No WAR race.
        if (k + 2 < NTILE) {
            async_tile16(cur, orow + (k + 2) * TT);
        }
    }
}

// ---------------------------------------------------------------------------
extern "C" void kernel_launch(void* const* d_in, const int* in_sizes, int n_in,
                              void* d_out, int out_size, void* d_ws, size_t ws_size,
                              hipStream_t stream)
{
    (void)in_sizes; (void)n_in; (void)out_size; (void)d_ws; (void)ws_size;
    const float* x  = (const float*)d_in[0];
    // d_in[1] = targets (unused by the reference outputs)
    const float* bw = (const float*)d_in[2];
    float* out = (float*)d_out;

    minmax_norm_kernel<<<NSEQ, 256, 0, stream>>>(x, out);      // writes origin half
    bsa_scan_kernel<<<NSEQ / 256, 256, 0, stream>>>(bw, out);  // writes decoded half
}